// SDRGNet_70549132804389
// MI455X (gfx1250) — compile-verified
//
#include <hip/hip_runtime.h>
#include <hip/hip_bf16.h>

typedef float v2f __attribute__((ext_vector_type(2)));
typedef float v8f __attribute__((ext_vector_type(8)));

#define MASK_FILL (-1e9f)
#define FLT_BIG   (3.402823466e+38f)

// relu in one VALU op: v_med3_num_f32(u, 0, +FLT_MAX)
__device__ __forceinline__ float relu1(float u) {
    return __builtin_amdgcn_fmed3f(u, 0.0f, FLT_BIG);
}

// ---------------------------------------------------------------------------
// Prep kernel: fold the post-ReLU linear layers.
//   w[k] = sum_n W2[k][n] * Ws[n]          (64 floats)
//   c    = sum_n b2[n]   * Ws[n] + bs      (1 float)
// wc[0..63] = w, wc[64] = c
// ---------------------------------------------------------------------------
__global__ void sdrgnet_prep(const float* __restrict__ W2,
                             const float* __restrict__ b2,
                             const float* __restrict__ Ws,
                             const float* __restrict__ bs,
                             float* __restrict__ wc) {
    int j = threadIdx.x;  // 0..63
    float acc = 0.f;
#pragma unroll
    for (int n = 0; n < 64; ++n) acc += W2[j * 64 + n] * Ws[n];
    wc[j] = acc;
    if (j == 0) {
        float c = bs[0];
        for (int n = 0; n < 64; ++n) c += b2[n] * Ws[n];
        wc[64] = c;
    }
}

// ---------------------------------------------------------------------------
// Main kernel: one wave handles 32 edges per iteration (two 16-edge tiles)
// using V_WMMA_F32_16X16X4_F32:
//   Ut(16 feat x 16 edges) = W1t_tile(16x4) x In4t(4x16) + b1_tile
// then score = sum_f w[f]*relu(Ut[f][e]) + c, masked, stored (full-wave 128B).
//
// A layout (32-bit A 16x4):  lanes 0-15: M=lane,    v0=K0, v1=K1
//                            lanes 16-31: M=lane-16, v0=K2, v1=K3
// B layout (4x16, mirrored): lanes 0-15: N=lane,    v0=K0 row, v1=K1 row
//                            lanes 16-31: N=lane-16, v0=K2 row, v1=K3 row
// C/D layout (16x16 f32): VGPR r: M = r (lanes 0-15) / 8+r (lanes 16-31), N = lane%16
// ---------------------------------------------------------------------------
__global__ void __launch_bounds__(256)
sdrgnet_edge_score(const float* __restrict__ x,                // [N_NODES, 1]
                   const int* __restrict__ edge_index,         // [2, E]
                   const float* __restrict__ edge_attr,        // [E, 2]
                   const unsigned char* __restrict__ edge_mask,// [E] (bool)
                   const float* __restrict__ W1,               // [4, 64] row-major
                   const float* __restrict__ b1,               // [64]
                   const float* __restrict__ wc,               // [65] from prep
                   float* __restrict__ out,                    // [E]
                   int E) {
    const int lane   = threadIdx.x & 31;
    const int laneLo = lane & 15;
    const bool hi    = lane >= 16;

    const int wavesPerBlock = blockDim.x >> 5;
    const int waveId  = blockIdx.x * wavesPerBlock + (threadIdx.x >> 5);
    const int nWaves  = gridDim.x * wavesPerBlock;
    const int nGroups = (E + 31) >> 5;   // 32 edges per group

    // ---- per-lane constants (all tiny, L2-hot) ----
    // A matrices: W1^T feature tiles. A[M][K] = W1[K][16*t + M].
    v2f A[4];
    {
        const int k0 = hi ? 2 : 0;
        const int k1 = hi ? 3 : 1;
#pragma unroll
        for (int t = 0; t < 4; ++t) {
            const int f = 16 * t + laneLo;
            A[t].x = W1[k0 * 64 + f];
            A[t].y = W1[k1 * 64 + f];
        }
    }
    // C-layout keyed constants: feature f(t,r,half) = 16t + r + (hi?8:0)
    float bias[4][8], wgt[4][8];
#pragma unroll
    for (int t = 0; t < 4; ++t)
#pragma unroll
        for (int r = 0; r < 8; ++r) {
            const int f = 16 * t + r + (hi ? 8 : 0);
            bias[t][r] = b1[f];
            wgt[t][r]  = wc[f];
        }
    const float cconst = wc[64];

    // ---- grid-stride over 32-edge groups ----
    for (int g = waveId; g < nGroups; g += nWaves) {
        const int e  = g * 32 + lane;
        const int ec = (e < E) ? e : (E - 1);   // clamped for safe loads

        // Uniform, fully-coalesced loads: every lane loads its own edge's data.
        const int   r  = edge_index[ec];        // row index array  (128B/wave)
        const int   c  = edge_index[E + ec];    // col index array  (128B/wave)
        const float xr = x[r];                  // gather (x is L2-resident)
        const float xc = x[c];                  // gather
        const float2 ea = *(const float2*)(edge_attr + 2 * ec); // 256B/wave

        // Cross-half swaps to build the two B matrices (4x16 in4^T).
        const float e0s = __shfl_xor(ea.x, 16, 32);
        const float e1s = __shfl_xor(ea.y, 16, 32);
        const float xrs = __shfl_xor(xr,   16, 32);
        const float xcs = __shfl_xor(xc,   16, 32);

        v2f BA, BB;   // tile A: edges g*32..+15, tile B: edges g*32+16..+31
        BA.x = hi ? e0s  : xr;   BA.y = hi ? e1s  : xc;
        BB.x = hi ? ea.x : xrs;  BB.y = hi ? ea.y : xcs;

        float accA = 0.f, accB = 0.f;
#pragma unroll
        for (int t = 0; t < 4; ++t) {
            v8f cin;
#pragma unroll
            for (int rr = 0; rr < 8; ++rr) cin[rr] = bias[t][rr];
            // D = A x B + C   (C carries the b1 bias) -> v_wmma_f32_16x16x4_f32
            v8f uA = __builtin_amdgcn_wmma_f32_16x16x4_f32(
                false, A[t], false, BA, (short)0, cin, false, false);
            v8f uB = __builtin_amdgcn_wmma_f32_16x16x4_f32(
                false, A[t], false, BB, (short)0, cin, false, false);
#pragma unroll
            for (int rr = 0; rr < 8; ++rr) {
                accA = fmaf(relu1(uA[rr]), wgt[t][rr], accA);
                accB = fmaf(relu1(uB[rr]), wgt[t][rr], accB);
            }
        }

        // Fold half-wave feature groups, then merge the two tiles so all 32
        // lanes hold the score of edge g*32+lane  -> one 128B coalesced store.
        accA += __shfl_xor(accA, 16, 32);
        accB += __shfl_xor(accB, 16, 32);
        const float s = (hi ? accB : accA) + cconst;

        if (e < E) out[e] = edge_mask[e] ? s : MASK_FILL;
    }
}

// ---------------------------------------------------------------------------
// Launch
// ---------------------------------------------------------------------------
extern "C" void kernel_launch(void* const* d_in, const int* in_sizes, int n_in,
                              void* d_out, int out_size, void* d_ws, size_t ws_size,
                              hipStream_t stream) {
    const float*         x          = (const float*)d_in[0];
    const int*           edge_index = (const int*)d_in[1];
    const float*         edge_attr  = (const float*)d_in[2];
    const unsigned char* edge_mask  = (const unsigned char*)d_in[3];
    const float*         W1         = (const float*)d_in[4];
    const float*         b1         = (const float*)d_in[5];
    const float*         W2         = (const float*)d_in[6];
    const float*         b2         = (const float*)d_in[7];
    const float*         Ws         = (const float*)d_in[8];
    const float*         bs         = (const float*)d_in[9];

    const int E = in_sizes[3];      // edge_mask element count == N_EDGES
    float* wc = (float*)d_ws;       // 65 floats of scratch

    sdrgnet_prep<<<1, 64, 0, stream>>>(W2, b2, Ws, bs, wc);

    // 32 edges per wave-iteration; 8 waves per 256-thread block; grid-stride.
    const int nGroups = (E + 31) / 32;
    int blocks = (nGroups + 7) / 8;
    if (blocks > 4096) blocks = 4096;   // grid-stride loop covers the rest
    if (blocks < 1) blocks = 1;

    sdrgnet_edge_score<<<blocks, 256, 0, stream>>>(
        x, edge_index, edge_attr, edge_mask, W1, b1, wc, (float*)d_out, E);
}